// VectorQuantizer_17377437679659
// MI455X (gfx1250) — compile-verified
//
#include <hip/hip_runtime.h>
#include <hip/hip_bf16.h>

typedef __attribute__((ext_vector_type(2))) float v2f;
typedef __attribute__((ext_vector_type(8))) float v8f;

#define VQ_B 16
#define VQ_N 1024
#define VQ_C 64
#define VQ_K 4096
#define VQ_BCN (VQ_B * VQ_N * VQ_C)   // 1,048,576

// ---------------------------------------------------------------- init: zero f_hat + loss accumulator
__global__ __launch_bounds__(256) void vq_init(float* __restrict__ fhat,
                                               float* __restrict__ acc) {
  int gid = blockIdx.x * 256 + threadIdx.x;
  fhat[gid] = 0.0f;
  if (gid == 0) *acc = 0.0f;
}

// ---------------------------------------------------------------- w_sq[k] = sum_c W[k,c]^2  (one wave per row)
__global__ __launch_bounds__(256) void vq_wsq(const float* __restrict__ W,
                                              float* __restrict__ wsq) {
  int wave = threadIdx.x >> 5;
  int lane = threadIdx.x & 31;
  int k = blockIdx.x * 8 + wave;           // 512 blocks * 8 waves = 4096 rows
  const float* wr = W + (size_t)k * VQ_C;
  float a = wr[lane];
  float b = wr[lane + 32];
  float s = a * a + b * b;
#pragma unroll
  for (int m = 16; m >= 1; m >>= 1) s += __shfl_xor(s, m, 32);
  if (lane == 0) wsq[k] = s;
}

// ---------------------------------------------------------------- pack codebook into WMMA B-fragment order:
// Wt[gid] with gid = chunk*512 + k*32 + lane  holds  W[chunk*16 + (lane&15)][4k + 2*(lane>>4) + {0,1}]
// -> inner-loop B loads become fully coalesced b64 at stride 256 B.
__global__ __launch_bounds__(256) void vq_pack(const float* __restrict__ W,
                                               v2f* __restrict__ Wt) {
  int gid   = blockIdx.x * 256 + threadIdx.x;   // 0 .. K*C/2 - 1 = 131071
  int lane  = gid & 31;
  int k     = (gid >> 5) & 15;
  int chunk = gid >> 9;
  int half  = lane >> 4;
  int l     = lane & 15;
  int code  = chunk * 16 + l;
  int koff  = 4 * k + 2 * half;
  v2f v;
  v.x = W[(size_t)code * VQ_C + koff + 0];
  v.y = W[(size_t)code * VQ_C + koff + 1];
  Wt[gid] = v;
}

// ---------------------------------------------------------------- area downsample of residual (f - f_hat) -> z [B*pn, 64]
__global__ __launch_bounds__(256) void vq_down(const float* __restrict__ f,
                                               const float* __restrict__ fhat,
                                               float* __restrict__ z, int pn) {
  int gid = blockIdx.x * 256 + threadIdx.x;
  int total = VQ_B * pn * VQ_C;
  if (gid >= total) return;
  int c = gid & (VQ_C - 1);
  int p = (gid >> 6) % pn;
  int b = gid / (VQ_C * pn);
  int S = VQ_N / pn;
  size_t base = ((size_t)b * VQ_N + (size_t)p * S) * VQ_C + c;
  float s = 0.0f;
  for (int j = 0; j < S; ++j) {
    size_t o = base + (size_t)j * VQ_C;
    s += f[o] - fhat[o];
  }
  z[gid] = s * (1.0f / (float)S);
}

// ---------------------------------------------------------------- WMMA f32 distance + argmin
// 4 waves per block; wave w scans codes [w*1024, (w+1)*1024) in 16-code chunks,
// 16 chained v_wmma_f32_16x16x4_f32 per chunk (K=64). Per-lane running argmin on
// the 8 C-slots (slot j -> row M = j+8*half, col N = chunk_base + (lane&15)),
// 16-lane xor-butterfly per half-wave, then LDS combine across the 4 waves
// (contiguous code ranges + index tie-break == global first-min, matching jnp.argmin).
__global__ __launch_bounds__(128) void vq_argmin_wmma(const float* __restrict__ z,
                                                      const v2f* __restrict__ Wt,
                                                      const float* __restrict__ wsq,
                                                      int* __restrict__ idx_out) {
  __shared__ float sval[4][16];
  __shared__ int   sidx[4][16];
  const int tid  = threadIdx.x;
  const int wave = tid >> 5;
  const int lane = tid & 31;
  const int half = lane >> 4;      // 0: K offsets 0,1   1: K offsets 2,3
  const int l    = lane & 15;
  const int row0 = blockIdx.x * 16;

  // A fragments: lane holds z[row0+l][4k + 2*half + {0,1}]
  const float* zr = z + ((size_t)(row0 + l)) * VQ_C + half * 2;
  v2f a[16];
#pragma unroll
  for (int k = 0; k < 16; ++k) {
    a[k].x = zr[4 * k + 0];
    a[k].y = zr[4 * k + 1];
  }

  float bestv[8];
  int   besti[8];
#pragma unroll
  for (int j = 0; j < 8; ++j) { bestv[j] = 3.402823466e38f; besti[j] = 0; }

  const int cb0 = wave * (VQ_K / 4);
  for (int cb = cb0; cb < cb0 + VQ_K / 4; cb += 16) {
    const v2f* wt = Wt + ((size_t)(cb >> 4)) * 512 + lane;   // coalesced
    v8f acc = {0.0f, 0.0f, 0.0f, 0.0f, 0.0f, 0.0f, 0.0f, 0.0f};
#pragma unroll
    for (int k = 0; k < 16; ++k) {
      v2f bfrag = wt[k * 32];
      acc = __builtin_amdgcn_wmma_f32_16x16x4_f32(
          false, a[k], false, bfrag, (short)0, acc, false, false);
    }
    float ws = wsq[cb + l];
    int col = cb + l;
#pragma unroll
    for (int j = 0; j < 8; ++j) {
      float s = ws - 2.0f * acc[j];
      if (s < bestv[j]) { bestv[j] = s; besti[j] = col; }
    }
  }

  // reduce (min, first-index) across the 16 lanes of each half-wave
#pragma unroll
  for (int m = 1; m <= 8; m <<= 1) {
#pragma unroll
    for (int j = 0; j < 8; ++j) {
      float ov = __shfl_xor(bestv[j], m, 32);
      int   oi = __shfl_xor(besti[j], m, 32);
      if (ov < bestv[j] || (ov == bestv[j] && oi < besti[j])) {
        bestv[j] = ov; besti[j] = oi;
      }
    }
  }
  if (l == 0) {
#pragma unroll
    for (int j = 0; j < 8; ++j) {
      sval[wave][half * 8 + j] = bestv[j];
      sidx[wave][half * 8 + j] = besti[j];
    }
  }
  __syncthreads();

  // combine the 4 waves' candidates per row
  if (tid < 16) {
    float bv = sval[0][tid];
    int   bi = sidx[0][tid];
#pragma unroll
    for (int w = 1; w < 4; ++w) {
      float ov = sval[w][tid];
      int   oi = sidx[w][tid];
      if (ov < bv || (ov == bv && oi < bi)) { bv = ov; bi = oi; }
    }
    idx_out[row0 + tid] = bi;
  }
}

// ---------------------------------------------------------------- gather codes, linear upsample, update f_hat,
// accumulate sum((f - f_hat)^2), optionally emit final output [B,N,C]
__global__ __launch_bounds__(256) void vq_update(const float* __restrict__ f,
                                                 float* __restrict__ fhat,
                                                 const float* __restrict__ W,
                                                 const int* __restrict__ idx,
                                                 float* __restrict__ out,
                                                 float* __restrict__ acc,
                                                 int pn, int last) {
  __shared__ float sred[256];
  int gid = blockIdx.x * 256 + threadIdx.x;        // B*N*C exactly, c innermost
  int c = gid & (VQ_C - 1);
  int n = (gid >> 6) & (VQ_N - 1);
  int b = gid >> 16;

  // F.interpolate(mode='linear', align_corners=False) coords, pn -> N
  float src = ((float)n + 0.5f) * ((float)pn / (float)VQ_N) - 0.5f;
  src = fmaxf(src, 0.0f);
  int i0 = (int)floorf(src);
  i0 = min(i0, pn - 1);
  int i1 = min(i0 + 1, pn - 1);
  float w1 = src - (float)i0;

  const int* ib = idx + b * pn;
  float h = W[(size_t)ib[i0] * VQ_C + c] * (1.0f - w1)
          + W[(size_t)ib[i1] * VQ_C + c] * w1;

  float fh = fhat[gid] + h;
  fhat[gid] = fh;
  float r = f[gid] - fh;
  if (last) out[gid] = fh;

  sred[threadIdx.x] = r * r;
  __syncthreads();
#pragma unroll
  for (int s = 128; s > 0; s >>= 1) {
    if (threadIdx.x < s) sred[threadIdx.x] += sred[threadIdx.x + s];
    __syncthreads();
  }
  if (threadIdx.x == 0) atomicAdd(acc, sred[0]);
}

// ---------------------------------------------------------------- finalize the two scalars
__global__ void vq_fin(const float* __restrict__ acc, float* __restrict__ out) {
  float s = *acc;
  const float inv = 1.0f / (11.0f * (float)VQ_BCN);
  out[VQ_BCN + 0] = 0.25f * s * inv;  // commit
  out[VQ_BCN + 1] = s * inv;          // qlat
}

extern "C" void kernel_launch(void* const* d_in, const int* in_sizes, int n_in,
                              void* d_out, int out_size, void* d_ws, size_t ws_size,
                              hipStream_t stream) {
  (void)in_sizes; (void)n_in; (void)out_size; (void)ws_size;
  const float* f = (const float*)d_in[0];   // [B,N,C] = [16,1024,64]
  const float* W = (const float*)d_in[1];   // [K,C]   = [4096,64]
  float* out = (float*)d_out;               // [B,N,C] ++ commit ++ qlat

  char* ws = (char*)d_ws;
  float* fhat = (float*)(ws);                               // 4 MB
  float* z    = (float*)(ws + (4u << 20));                  // 4 MB (max B*1024*64)
  v2f*   Wt   = (v2f*)  (ws + (8u << 20));                  // 1 MB packed codebook
  float* wsq  = (float*)(ws + (9u << 20));                  // 16 KB
  int*   idx  = (int*)  (ws + (9u << 20) + (64u << 10));    // 64 KB
  float* acc  = (float*)(ws + (9u << 20) + (192u << 10));   // 4 B

  vq_init<<<VQ_BCN / 256, 256, 0, stream>>>(fhat, acc);
  vq_wsq<<<VQ_K / 8, 256, 0, stream>>>(W, wsq);
  vq_pack<<<(VQ_K * VQ_C / 2) / 256, 256, 0, stream>>>(W, Wt);

  static const int scales[11] = {1, 2, 4, 8, 16, 32, 64, 128, 256, 512, 1024};
  for (int si = 0; si < 11; ++si) {
    int pn = scales[si];
    int dtotal = VQ_B * pn * VQ_C;
    vq_down<<<(dtotal + 255) / 256, 256, 0, stream>>>(f, fhat, z, pn);
    vq_argmin_wmma<<<(VQ_B * pn) / 16, 128, 0, stream>>>(z, Wt, wsq, idx);
    vq_update<<<VQ_BCN / 256, 256, 0, stream>>>(f, fhat, W, idx, out, acc,
                                                pn, si == 10 ? 1 : 0);
  }
  vq_fin<<<1, 1, 0, stream>>>(acc, out);
}